// ODEFunc_11141145166257
// MI455X (gfx1250) — compile-verified
//
#include <hip/hip_runtime.h>

// CDNA5 / gfx1250: fp32 WMMA 16x16x4 fused CNF ODE-func kernel.
// D = A(16x4) x B(4x16) + C(16x16), wave32, matrices striped per ISA 7.12.2:
//   A: lane = (k>>1)*16 + m ... concretely vgpr j holds A[m=lane&15][k=4*kk+2*(lane>>4)+j]
//   B: vgpr j holds B[k=4*kk+2*(lane>>4)+j][n=lane&15]
//   C/D: vgpr v holds C[m=v+8*(lane>>4)][n=lane&15]

typedef __attribute__((ext_vector_type(2))) float v2f;
typedef __attribute__((ext_vector_type(8))) float v8f;

__global__ __launch_bounds__(256) void odefunc_cnf_kernel(
    const float* __restrict__ zlp,   // B x 3 (z0, z1, logp)
    const float* __restrict__ W1,    // 64 x 2
    const float* __restrict__ b1,    // 64
    const float* __restrict__ W2,    // 64 x 64
    const float* __restrict__ b2,    // 64
    const float* __restrict__ W3,    // 2 x 64
    const float* __restrict__ b3,    // 2
    float* __restrict__ out,         // B x 3 (dz0, dz1, -trace)
    int nTiles, int B)
{
    // W2 staged in WMMA-B register layout:
    // sW2B[nt*1024 + kk*64 + h*32 + l*2 + j] = W2[16*nt + l][4*kk + 2*h + j]
    __shared__ float sW2B[4096];
    __shared__ float sW1x[64], sW1y[64], sb1[64], sb2[64];
    __shared__ float sW3[128];
    __shared__ float sb3[2];

    const int tid = threadIdx.x;

    #pragma unroll
    for (int it = 0; it < 16; ++it) {
        int i  = it * 256 + tid;
        int j  = i & 1;
        int l  = (i >> 1) & 15;
        int h  = (i >> 5) & 1;
        int kk = (i >> 6) & 15;
        int nt = i >> 10;
        sW2B[i] = W2[(16 * nt + l) * 64 + 4 * kk + 2 * h + j];
    }
    if (tid < 64) {
        sW1x[tid] = W1[tid * 2 + 0];
        sW1y[tid] = W1[tid * 2 + 1];
        sb1[tid]  = b1[tid];
        sb2[tid]  = b2[tid];
    } else if (tid < 192) {
        sW3[tid - 64] = W3[tid - 64];
    } else if (tid < 194) {
        sb3[tid - 192] = b3[tid - 192];
    }
    __syncthreads();

    const int wave = tid >> 5;
    const int lane = tid & 31;
    const int half = lane >> 4;   // which 16-lane half
    const int l15  = lane & 15;
    const int tile = blockIdx.x * 8 + wave;
    if (tile >= nTiles) return;   // wave-uniform: EXEC all-ones below

    const int row0 = tile * 16;
    int rl = row0 + l15;          // this lane's A-matrix row (sample)
    if (rl >= B) rl = B - 1;      // clamp (B is a multiple of 16 in practice)
    const float z0 = zlp[rl * 3 + 0];
    const float z1 = zlp[rl * 3 + 1];

    // Accumulators: primal (bias-initialized with b2) + two tangents, 4 N-tiles.
    v8f acc_h[4], acc_t0[4], acc_t1[4];
    #pragma unroll
    for (int nt = 0; nt < 4; ++nt) {
        float bias = sb2[nt * 16 + l15];   // depends on N only
        #pragma unroll
        for (int v = 0; v < 8; ++v) {
            acc_h[nt][v]  = bias;
            acc_t0[nt][v] = 0.0f;
            acc_t1[nt][v] = 0.0f;
        }
    }

    // K loop over layer-1 features in chunks of 4.
    #pragma unroll
    for (int kk = 0; kk < 16; ++kk) {
        v2f aH, aT0, aT1;
        #pragma unroll
        for (int j = 0; j < 2; ++j) {
            int c = 4 * kk + 2 * half + j;         // half-uniform -> LDS broadcast
            float w0 = sW1x[c], w1 = sW1y[c];
            float a  = fmaf(z0, w0, fmaf(z1, w1, sb1[c]));
            float e  = __expf(a);
            float hh = a > 0.0f ? a    : e - 1.0f; // elu(a)
            float ep = a > 0.0f ? 1.0f : e;        // elu'(a)
            aH[j]  = hh;
            aT0[j] = ep * w0;                      // tangent e0 through layer 1
            aT1[j] = ep * w1;                      // tangent e1 through layer 1
        }
        #pragma unroll
        for (int nt = 0; nt < 4; ++nt) {
            int bidx = nt * 1024 + kk * 64 + half * 32 + l15 * 2;
            v2f bm;
            bm.x = sW2B[bidx];
            bm.y = sW2B[bidx + 1];
            acc_h[nt]  = __builtin_amdgcn_wmma_f32_16x16x4_f32(
                false, aH,  false, bm, (short)0, acc_h[nt],  false, false);
            acc_t0[nt] = __builtin_amdgcn_wmma_f32_16x16x4_f32(
                false, aT0, false, bm, (short)0, acc_t0[nt], false, false);
            acc_t1[nt] = __builtin_amdgcn_wmma_f32_16x16x4_f32(
                false, aT1, false, bm, (short)0, acc_t1[nt], false, false);
        }
    }

    // Epilogue: layer-2 ELU + fold W3 (N-reduction) per lane.
    float p0[8], p1[8], pt[8];
    #pragma unroll
    for (int v = 0; v < 8; ++v) { p0[v] = 0.0f; p1[v] = 0.0f; pt[v] = 0.0f; }

    #pragma unroll
    for (int nt = 0; nt < 4; ++nt) {
        int n = nt * 16 + l15;
        float w30 = sW3[n], w31 = sW3[64 + n];
        #pragma unroll
        for (int v = 0; v < 8; ++v) {
            float a2  = acc_h[nt][v];
            float e2  = __expf(a2);
            float h2  = a2 > 0.0f ? a2   : e2 - 1.0f;
            float ep2 = a2 > 0.0f ? 1.0f : e2;
            p0[v] = fmaf(w30, h2, p0[v]);                  // out row 0
            p1[v] = fmaf(w31, h2, p1[v]);                  // out row 1
            pt[v] = fmaf(w30 * ep2, acc_t0[nt][v],         // J00 + J11
                    fmaf(w31 * ep2, acc_t1[nt][v], pt[v]));
        }
    }

    const float ob0 = sb3[0], ob1 = sb3[1];
    #pragma unroll
    for (int v = 0; v < 8; ++v) {
        float r0 = p0[v], r1 = p1[v], rt = pt[v];
        // reduce across the 16 lanes of this half (xor masks < 16 stay in-half)
        #pragma unroll
        for (int m = 1; m < 16; m <<= 1) {
            r0 += __shfl_xor(r0, m, 32);
            r1 += __shfl_xor(r1, m, 32);
            rt += __shfl_xor(rt, m, 32);
        }
        if (l15 == 0) {
            int grow = row0 + v + 8 * half;   // C-layout: M = v + 8*half
            if (grow < B) {
                out[grow * 3 + 0] = r0 + ob0;
                out[grow * 3 + 1] = r1 + ob1;
                out[grow * 3 + 2] = -rt;
            }
        }
    }
}

extern "C" void kernel_launch(void* const* d_in, const int* in_sizes, int n_in,
                              void* d_out, int out_size, void* d_ws, size_t ws_size,
                              hipStream_t stream) {
    // setup_inputs order: t, z_and_logp, W1, b1, W2, b2, W3, b3   (t is unused)
    const float* zlp = (const float*)d_in[1];
    const float* W1  = (const float*)d_in[2];
    const float* b1  = (const float*)d_in[3];
    const float* W2  = (const float*)d_in[4];
    const float* b2  = (const float*)d_in[5];
    const float* W3  = (const float*)d_in[6];
    const float* b3  = (const float*)d_in[7];

    const int B      = in_sizes[1] / 3;       // z_and_logp is B x 3
    const int nTiles = (B + 15) / 16;
    const int nBlk   = (nTiles + 7) / 8;      // 8 waves (tiles) per block

    odefunc_cnf_kernel<<<nBlk, 256, 0, stream>>>(
        zlp, W1, b1, W2, b2, W3, b3, (float*)d_out, nTiles, B);
}